// Conv_7524782702706
// MI455X (gfx1250) — compile-verified
//
#include <hip/hip_runtime.h>

typedef float v2f __attribute__((ext_vector_type(2)));
typedef float v8f __attribute__((ext_vector_type(8)));
typedef int   v4i __attribute__((ext_vector_type(4)));

#define IMG_H 256
#define IMG_W 256
#define TILE 16
#define NWAVES 8
#define BLOCK_COLS (NWAVES * TILE)   // 128 output columns per block
#define LDS_W 136                    // body at j=4..131 (16B aligned), halos at j=3 / j=132
#define LDS_H 20                     // 18 halo rows + 2 zero-tap rows for the last K=4 chunk

#if defined(__AMDGCN__) && __has_builtin(__builtin_amdgcn_global_load_async_to_lds_b128)
#define HAVE_ASYNC 1
#endif

__device__ __forceinline__ void wait_async_zero() {
#if __has_builtin(__builtin_amdgcn_s_wait_asynccnt)
  __builtin_amdgcn_s_wait_asynccnt(0);
#else
  asm volatile("s_wait_asynccnt 0" ::: "memory");
#endif
}

__global__ __launch_bounds__(256) void conv3x3_wmma_kernel(
    const float* __restrict__ X, const float* __restrict__ Kf,
    float* __restrict__ Y) {
  __shared__ float lds[LDS_H * LDS_W];

  const int rowGroups = IMG_H / TILE;        // 16
  const int colGroups = IMG_W / BLOCK_COLS;  // 2
  int bid   = blockIdx.x;
  int plane = bid / (rowGroups * colGroups);
  int rem   = bid - plane * (rowGroups * colGroups);
  int rg    = rem / colGroups;
  int cg    = rem - rg * colGroups;
  int row0  = rg * TILE;        // first output row of block
  int col0  = cg * BLOCK_COLS;  // first output col of block

  const float* __restrict__ src = X + (size_t)plane * (IMG_H * IMG_W);

  int tid  = threadIdx.x;
  int wave = tid >> 5;
  int lane = tid & 31;

  // 3x3 taps (uniform -> scalar loads)
  float kw[3][3];
#pragma unroll
  for (int i = 0; i < 3; ++i)
#pragma unroll
    for (int j = 0; j < 3; ++j)
      kw[i][j] = Kf[i * 3 + j];

  // ---- stage rows row0-1 .. row0+18 into LDS.
  // LDS row layout: j=3 <-> image col col0-1 (left halo), j=4..131 <-> cols col0..col0+127
  //                 j=132 <-> col col0+128 (right halo)
#ifdef HAVE_ASYNC
  {
    typedef __attribute__((address_space(1))) v4i gvec_t;
    typedef __attribute__((address_space(3))) v4i lvec_t;
    // Body: one async b128 per in-bounds row (wave-uniform predicate -> full EXEC on issue).
    for (int i = wave; i < LDS_H; i += NWAVES) {
      int r = row0 - 1 + i;
      if (r >= 0 && r < IMG_H) {
        gvec_t* g = (gvec_t*)(src + (size_t)r * IMG_W + col0 + lane * 4);
        lvec_t* l = (lvec_t*)&lds[i * LDS_W + 4 + lane * 4];
        __builtin_amdgcn_global_load_async_to_lds_b128(g, l, 0, 0);
      } else {
        float4 z = make_float4(0.f, 0.f, 0.f, 0.f);
        *(float4*)&lds[i * LDS_W + 4 + lane * 4] = z;
      }
    }
    // Halo columns (and their border zeros): 2 cells per row.
    if (tid < 2 * LDS_H) {
      int side = tid & 1;
      int i    = tid >> 1;
      int r = row0 - 1 + i;
      int c = side ? (col0 + BLOCK_COLS) : (col0 - 1);
      float v = 0.f;
      if (r >= 0 && r < IMG_H && c >= 0 && c < IMG_W) v = src[r * IMG_W + c];
      lds[i * LDS_W + (side ? 4 + BLOCK_COLS : 3)] = v;
    }
    wait_async_zero();
  }
#else
  {
    const int STAGE_COLS = BLOCK_COLS + 2;       // 130
    const int STAGE_N    = LDS_H * STAGE_COLS;   // 2600
    for (int idx = tid; idx < STAGE_N; idx += 256) {
      int i = idx / STAGE_COLS;
      int j = idx - i * STAGE_COLS;
      int r = row0 - 1 + i;
      int c = col0 - 1 + j;
      int rc = min(max(r, 0), IMG_H - 1);
      int cc = min(max(c, 0), IMG_W - 1);
      float v = src[rc * IMG_W + cc];
      bool inb = (r >= 0) & (r < IMG_H) & (c >= 0) & (c < IMG_W);
      lds[i * LDS_W + 3 + j] = inb ? v : 0.0f;
    }
  }
#endif
  __syncthreads();

  int half = lane >> 4;   // 0: lanes 0-15, 1: lanes 16-31
  int mn   = lane & 15;   // A-row (M) for this lane, also B/D column (N)

  v8f acc = {0.f, 0.f, 0.f, 0.f, 0.f, 0.f, 0.f, 0.f};

  // O[m][n] = sum_kx sum_k T_kx[m][k] * lds[k][3 + wave*16 + n + kx],
  // T_kx[m][k] = kw[k-m][kx] for (k-m) in [0,2], else 0. K-dim = 18 rows padded to 20.
#pragma unroll
  for (int kx = 0; kx < 3; ++kx) {
    float t0 = kw[0][kx], t1 = kw[1][kx], t2 = kw[2][kx];
#pragma unroll
    for (int c4 = 0; c4 < 5; ++c4) {
      int kb = 4 * c4 + half * 2;  // this lane's K index for A.x/B.x; +1 for A.y/B.y
      // A operand: 16x4 Toeplitz chunk (lanes 0-15 hold chunk K0,K1; lanes 16-31 K2,K3)
      int d0 = kb - mn;
      int d1 = d0 + 1;
      v2f A;
      A.x = (d0 == 0) ? t0 : (d0 == 1) ? t1 : (d0 == 2) ? t2 : 0.0f;
      A.y = (d1 == 0) ? t0 : (d1 == 1) ? t1 : (d1 == 2) ? t2 : 0.0f;
      // B operand: 4x16 input chunk, row-striped across lanes (VGPR0: rows kb/kb+2, VGPR1: kb+1/kb+3)
      int jb = 3 + wave * TILE + mn + kx;
      v2f Bv;
      Bv.x = lds[kb * LDS_W + jb];
      Bv.y = lds[(kb + 1) * LDS_W + jb];
      acc = __builtin_amdgcn_wmma_f32_16x16x4_f32(false, A, false, Bv,
                                                  (short)0, acc, false, false);
    }
  }

  // ---- store D: VGPR v -> output row row0 + v + 8*half, col col0 + wave*16 + n
  float* __restrict__ dst = Y + (size_t)plane * (IMG_H * IMG_W);
  int ocol = col0 + wave * TILE + mn;
  int orow = row0 + half * 8;
#pragma unroll
  for (int v = 0; v < 8; ++v) {
    dst[(orow + v) * IMG_W + ocol] = acc[v];
  }
}

extern "C" void kernel_launch(void* const* d_in, const int* in_sizes, int n_in,
                              void* d_out, int out_size, void* d_ws, size_t ws_size,
                              hipStream_t stream) {
  const float* X = (const float*)d_in[0];
  const float* K = (const float*)d_in[1];
  float* Y = (float*)d_out;
  int planes = in_sizes[0] / (IMG_H * IMG_W);  // 16*64 = 1024
  dim3 grid((unsigned)(planes * (IMG_H / TILE) * (IMG_W / BLOCK_COLS)));
  conv3x3_wmma_kernel<<<grid, dim3(256), 0, stream>>>(X, K, Y);
}